// CSCLoss_20134806684275
// MI455X (gfx1250) — compile-verified
//
#include <hip/hip_runtime.h>
#include <hip/hip_bf16.h>
#include <math.h>

typedef __attribute__((ext_vector_type(2))) float v2f;
typedef __attribute__((ext_vector_type(8))) float v8f;

#define BB     32
#define NN     32
#define CC     256
#define NPAIR  6
#define GROUPS 64      /* B*N / 16 points per group */
#define EPSN   1e-12f

// One workgroup = 16 sample points. Fill S[level][point][channel] in LDS with
// fp32 bilinear samples, then wave 0 computes diag(S_l * S_m^T) for the 6
// (l,m) pairs via V_WMMA_F32_16X16X4_F32 accumulation over K=256.
__global__ __launch_bounds__(256) void csc_sample_gram(
    const float* __restrict__ f0, const float* __restrict__ f1,
    const float* __restrict__ f2, const float* __restrict__ boxes,
    float* __restrict__ partial)
{
    __shared__ float S[3][16][CC];        // 48 KB
    __shared__ float dg[NPAIR][16];       // per-pair Gram diagonals

    const int g   = blockIdx.x;           // point-group id
    const int tid = threadIdx.x;          // 0..255
    const int pl  = tid >> 4;             // point-in-group 0..15
    const int ci  = tid & 15;             // channel slice 0..15

    const int P = g * 16 + pl;            // global point id = b*N + n
    const int b = P / NN;

    const float cx = boxes[(size_t)P * 4 + 0];
    const float cy = boxes[(size_t)P * 4 + 1];

    const float* feats[3] = { f0, f1, f2 };
    const int    dims[3]  = { 64, 32, 16 };

    #pragma unroll
    for (int l = 0; l < 3; ++l) {
        const int W = dims[l], H = dims[l];
        const int HW = H * W;
        // align_corners=True + border padding: pixel = clamp(cx*(W-1), 0, W-1)
        float px = fminf(fmaxf(cx * (float)(W - 1), 0.f), (float)(W - 1));
        float py = fminf(fmaxf(cy * (float)(H - 1), 0.f), (float)(H - 1));
        int x0 = (int)floorf(px);
        int y0 = (int)floorf(py);
        int x1 = min(x0 + 1, W - 1);
        int y1 = min(y0 + 1, H - 1);
        float wx = px - (float)x0;
        float wy = py - (float)y0;
        float w00 = (1.f - wx) * (1.f - wy);
        float w01 = wx * (1.f - wy);
        float w10 = (1.f - wx) * wy;
        float w11 = wx * wy;
        const float* fb = feats[l] + (size_t)b * CC * HW;
        const size_t o00 = (size_t)y0 * W + x0;
        const size_t o01 = (size_t)y0 * W + x1;
        const size_t o10 = (size_t)y1 * W + x0;
        const size_t o11 = (size_t)y1 * W + x1;
        #pragma unroll 4
        for (int j = 0; j < 16; ++j) {
            int c = ci * 16 + j;
            const float* fc = fb + (size_t)c * HW;
            S[l][pl][c] = fc[o00]*w00 + fc[o01]*w01 + fc[o10]*w10 + fc[o11]*w11;
        }
    }
    __syncthreads();

    if (tid < 32) {                       // wave 0, EXEC all ones
        const int lane = tid;
        const int p    = lane & 15;       // row (M for A, N for B)
        const int hi   = lane >> 4;       // K half-select
        const int pairs[NPAIR][2] = { {0,0},{1,1},{2,2},{0,1},{0,2},{1,2} };
        #pragma unroll
        for (int q = 0; q < NPAIR; ++q) {
            const int l = pairs[q][0];
            const int m = pairs[q][1];
            v8f acc = {};
            for (int k = 0; k < CC; k += 4) {
                // fp32 16x4 A layout: lane = M, VGPR0/1 = K {0,1} (+2 for lanes 16-31)
                // fp32 4x16 B layout mirrors it with lane = N.
                v2f av, bv;
                av.x = S[l][p][k + 2*hi];
                av.y = S[l][p][k + 2*hi + 1];
                bv.x = S[m][p][k + 2*hi];
                bv.y = S[m][p][k + 2*hi + 1];
                acc = __builtin_amdgcn_wmma_f32_16x16x4_f32(
                          /*neg_a=*/false, av, /*neg_b=*/false, bv,
                          /*c_mod=*/(short)0, acc,
                          /*reuse_a=*/false, /*reuse_b=*/false);
            }
            // D layout: VGPR r, lanes 0-15 -> (M=r, N=lane); lanes 16-31 -> (M=r+8, N=lane-16)
            // diag(d): d<8 at (VGPR d, lane d); d>=8 at (VGPR d-8, lane d+16)
            #pragma unroll
            for (int r = 0; r < 8; ++r) {
                if (lane == r)      dg[q][r]     = acc[r];
                if (lane == r + 24) dg[q][r + 8] = acc[r];
            }
        }
    }
    __syncthreads();

    if (tid == 0) {
        float total = 0.f;
        #pragma unroll
        for (int p = 0; p < 16; ++p) {
            float n0 = fmaxf(sqrtf(dg[0][p]), EPSN);
            float n1 = fmaxf(sqrtf(dg[1][p]), EPSN);
            float n2 = fmaxf(sqrtf(dg[2][p]), EPSN);
            total += dg[3][p] / (n0 * n1)
                   + dg[4][p] / (n0 * n2)
                   + dg[5][p] / (n1 * n2);
        }
        partial[g] = total;              // deterministic per-group partial
    }
}

__global__ void csc_finalize(const float* __restrict__ partial,
                             float* __restrict__ out)
{
    if (threadIdx.x == 0 && blockIdx.x == 0) {
        float total = 0.f;
        for (int g = 0; g < GROUPS; ++g) total += partial[g];  // fixed order
        float avg  = total / (float)(BB * NN * 3);
        float loss = 1.f - avg;
        if (isnan(loss))      loss = 0.f;                 // nan -> 0
        else if (isinf(loss)) loss = (loss > 0.f) ? 1.f : 0.f;  // +inf->1, -inf->0
        loss = fminf(fmaxf(loss, 0.f), 2.f);
        out[0] = loss;
    }
}

extern "C" void kernel_launch(void* const* d_in, const int* in_sizes, int n_in,
                              void* d_out, int out_size, void* d_ws, size_t ws_size,
                              hipStream_t stream)
{
    const float* f0    = (const float*)d_in[0];
    const float* f1    = (const float*)d_in[1];
    const float* f2    = (const float*)d_in[2];
    const float* boxes = (const float*)d_in[3];
    float* partial     = (float*)d_ws;       // 64 floats of scratch
    float* out         = (float*)d_out;

    csc_sample_gram<<<GROUPS, 256, 0, stream>>>(f0, f1, f2, boxes, partial);
    csc_finalize<<<1, 32, 0, stream>>>(partial, out);
}